// GraphTransformer_29472065585573
// MI455X (gfx1250) — compile-verified
//
#include <hip/hip_runtime.h>
#include <hip/hip_bf16.h>

typedef __attribute__((ext_vector_type(2))) float v2f;
typedef __attribute__((ext_vector_type(8))) float v8f;

// ---------------------------------------------------------------------------
// Generic fp32 WMMA GEMM:  C[n x M] = A[n x K] @ B[K x M] + bias (optional)
// grid.x = n/16 (n % 16 == 0), block = (32, 8). Each wave computes 16x16
// output tiles via V_WMMA_F32_16X16X4_F32, K-loop in steps of 4.
// A tile (16 x K) staged in LDS with padded stride (K+2) to avoid bank
// conflicts (natural stride K=128/512 would alias all 64 banks 16-way).
// ---------------------------------------------------------------------------
__global__ __launch_bounds__(256)
void gemm_wmma_f32(const float* __restrict__ A, int lda,
                   const float* __restrict__ B, int ldb,
                   const float* __restrict__ bias,
                   float* __restrict__ C, int ldc,
                   int K, int M)
{
    __shared__ float As[16 * 514];           // K <= 512, stride K+2
    const int Kp  = K + 2;
    const int r0  = blockIdx.x * 16;
    const int tid = threadIdx.y * 32 + threadIdx.x;

    for (int idx = tid; idx < 16 * K; idx += 256) {
        int rr = idx / K, cc = idx - rr * K;
        As[rr * Kp + cc] = A[(size_t)(r0 + rr) * lda + cc];
    }
    __syncthreads();

    const int lane = threadIdx.x;
    const int m    = lane & 15;              // row (A) / col (B,C) within tile
    const int kq   = (lane >> 4) << 1;       // K sub-offset: 0 or 2
    const int nTiles = M >> 4;

    for (int ct = threadIdx.y; ct < nTiles; ct += 8) {
        const int colBase = ct << 4;
        const float* Bp = B + colBase + m;
        v8f acc = {};
        for (int k = 0; k < K; k += 4) {
            v2f a, b;
            a.x = As[m * Kp + k + kq];
            a.y = As[m * Kp + k + kq + 1];
            b.x = Bp[(size_t)(k + kq) * ldb];
            b.y = Bp[(size_t)(k + kq + 1) * ldb];
            acc = __builtin_amdgcn_wmma_f32_16x16x4_f32(
                      false, a, false, b, (short)0, acc, false, false);
        }
        const float bv = bias ? bias[colBase + m] : 0.0f;
        const int rowHalf = (lane >> 4) << 3;     // 0 or 8
#pragma unroll
        for (int r = 0; r < 8; ++r) {
            C[(size_t)(r0 + rowHalf + r) * ldc + colBase + m] = acc[r] + bv;
        }
    }
}

// ---------------------------------------------------------------------------
// Utility fills
// ---------------------------------------------------------------------------
__global__ void fill_f32(float* p, float v, size_t n) {
    size_t i = (size_t)blockIdx.x * 256 + threadIdx.x;
    if (i < n) p[i] = v;
}
__global__ void fill_u32(unsigned* p, unsigned v, size_t n) {
    size_t i = (size_t)blockIdx.x * 256 + threadIdx.x;
    if (i < n) p[i] = v;
}
__global__ void copy_x_to_xcat(const float* __restrict__ x, float* __restrict__ xcat, size_t n128) {
    size_t i = (size_t)blockIdx.x * 256 + threadIdx.x;
    if (i >= n128) return;
    size_t node = i >> 7; int ch = (int)(i & 127);
    xcat[node * 512 + ch] = x[i];
}

// ---------------------------------------------------------------------------
// Degree: deg = segment_sum(1 over row) + 1 ; dis = deg^-0.5 ; invdeg = 1/deg
// ---------------------------------------------------------------------------
__global__ void deg_count(const int* __restrict__ row, float* deg, int E) {
    int e = blockIdx.x * 256 + threadIdx.x;
    if (e < E) atomicAdd(&deg[row[e]], 1.0f);
}
__global__ void deg_final(float* deg, float* dis, float* invdeg, int N) {
    int i = blockIdx.x * 256 + threadIdx.x;
    if (i >= N) return;
    float d = deg[i] + 1.0f;
    deg[i] = d; dis[i] = rsqrtf(d); invdeg[i] = 1.0f / d;
}

// ---------------------------------------------------------------------------
// GCN scatter, with the E x 16 @ 16 x 128 edge-feature GEMM fused in.
// 32 lanes per edge, 4 channels per lane (float4). We (8 KB) staged in LDS.
//   msg = dis[row]*dis[col] * relu(hw[row] + edge_attr@We + be)
//   agg[col] += msg   (float atomics, L2-resident)
// ---------------------------------------------------------------------------
__global__ __launch_bounds__(256)
void gcn_scatter(const float* __restrict__ hw, const float* __restrict__ edge_attr,
                 const float* __restrict__ We, const float* __restrict__ be,
                 const int* __restrict__ row, const int* __restrict__ col,
                 const float* __restrict__ dis, float* __restrict__ agg, int E)
{
    __shared__ float WeS[16 * 128];
    __shared__ float beS[128];
    const int tid = threadIdx.x;
    for (int i = tid; i < 16 * 128; i += 256) WeS[i] = We[i];
    if (tid < 128) beS[tid] = be[tid];
    __syncthreads();

    const int lane = tid & 31;
    const int e = blockIdx.x * 8 + (tid >> 5);
    if (e >= E) return;

    const int r = row[e], c = col[e];
    const float norm = dis[r] * dis[c];
    const int ch = lane * 4;

    float4 acc = *(const float4*)(beS + ch);
#pragma unroll
    for (int k = 0; k < 16; ++k) {
        float ea = edge_attr[(size_t)e * 16 + k];       // broadcast across wave
        float4 w = *(const float4*)(WeS + k * 128 + ch);
        acc.x += ea * w.x; acc.y += ea * w.y; acc.z += ea * w.z; acc.w += ea * w.w;
    }
    float4 hv = *(const float4*)(hw + (size_t)r * 128 + ch);
    float mx = norm * fmaxf(hv.x + acc.x, 0.0f);
    float my = norm * fmaxf(hv.y + acc.y, 0.0f);
    float mz = norm * fmaxf(hv.z + acc.z, 0.0f);
    float mw = norm * fmaxf(hv.w + acc.w, 0.0f);
    float* dst = agg + (size_t)c * 128 + ch;
    atomicAdd(dst + 0, mx); atomicAdd(dst + 1, my);
    atomicAdd(dst + 2, mz); atomicAdd(dst + 3, mw);
}

// h_next = relu(agg + relu(hw + root)/deg) -> written into xcat block column
__global__ void gcn_combine(const float* __restrict__ hw, const float* __restrict__ agg,
                            const float* __restrict__ root, const float* __restrict__ invdeg,
                            float* __restrict__ out, int ldc, size_t n128)
{
    size_t i = (size_t)blockIdx.x * 256 + threadIdx.x;
    if (i >= n128) return;
    size_t node = i >> 7; int ch = (int)(i & 127);
    float self = fmaxf(hw[i] + root[ch], 0.0f) * invdeg[node];
    out[node * (size_t)ldc + ch] = fmaxf(agg[i] + self, 0.0f);
}

// ---------------------------------------------------------------------------
// BatchNorm over N for 512 features: per-block LDS bins + global atomics
// ---------------------------------------------------------------------------
__global__ __launch_bounds__(256)
void bn_stats(const float* __restrict__ xcat, float* __restrict__ gsum,
              float* __restrict__ gsumsq, size_t total)
{
    __shared__ float ls[512], lsq[512];
    const int tid = threadIdx.x;
    for (int i = tid; i < 512; i += 256) { ls[i] = 0.f; lsq[i] = 0.f; }
    __syncthreads();
    for (size_t i = (size_t)blockIdx.x * 256 + tid; i < total; i += (size_t)gridDim.x * 256) {
        float v = xcat[i]; int c = (int)(i & 511);
        atomicAdd(&ls[c], v); atomicAdd(&lsq[c], v * v);
    }
    __syncthreads();
    for (int i = tid; i < 512; i += 256) {
        atomicAdd(&gsum[i], ls[i]); atomicAdd(&gsumsq[i], lsq[i]);
    }
}
__global__ void bn_final(const float* gsum, const float* gsumsq,
                         const float* gamma, const float* beta,
                         float* scale, float* shift, float Nf)
{
    int i = blockIdx.x * 256 + threadIdx.x;
    if (i >= 512) return;
    float mean = gsum[i] / Nf;
    float var  = gsumsq[i] / Nf - mean * mean;
    float sc   = gamma[i] * rsqrtf(var + 1e-5f);
    scale[i] = sc; shift[i] = beta[i] - mean * sc;
}
__global__ void bn_norm(float* __restrict__ xcat, const float* __restrict__ scale,
                        const float* __restrict__ shift, size_t total)
{
    size_t i = (size_t)blockIdx.x * 256 + threadIdx.x;
    if (i >= total) return;
    int c = (int)(i & 511);
    xcat[i] = xcat[i] * scale[c] + shift[c];
}

// ---------------------------------------------------------------------------
// Segment-softmax attention over dag edges. Monotone uint key for float max.
// ---------------------------------------------------------------------------
__device__ __forceinline__ unsigned f2key(float f) {
    unsigned u = __float_as_uint(f);
    return u ^ (unsigned)(((int)u >> 31) | 0x80000000);
}
__device__ __forceinline__ float key2f(unsigned u) {
    unsigned b = (u & 0x80000000u) ? (u ^ 0x80000000u) : ~u;
    return __uint_as_float(b);
}

// 8 threads/edge (one per head): logit = 0.25 * dot16(qk_src[src], qk_dst[dst])
__global__ __launch_bounds__(256)
void attn_logits(const float* __restrict__ qk, const int* __restrict__ src,
                 const int* __restrict__ dst, float* __restrict__ logits,
                 unsigned* __restrict__ m_u, int E)
{
    int tid = threadIdx.x;
    int e = blockIdx.x * 32 + (tid >> 3);
    int h = tid & 7;
    if (e >= E) return;
    int sj = src[e], di = dst[e];
    const float4* qs = (const float4*)(qk + (size_t)sj * 256 + h * 16);
    const float4* qd = (const float4*)(qk + (size_t)di * 256 + 128 + h * 16);
    float acc = 0.f;
#pragma unroll
    for (int t = 0; t < 4; ++t) {
        float4 a = qs[t], b = qd[t];
        acc += a.x * b.x + a.y * b.y + a.z * b.z + a.w * b.w;
    }
    acc *= 0.25f;                      // (D/H)^-0.5 = 16^-0.5
    logits[(size_t)e * 8 + h] = acc;
    atomicMax(&m_u[(size_t)di * 8 + h], f2key(acc));
}

// one thread per (edge, head): p = exp(logit - m[dst]); s[dst] += p
__global__ void attn_exp(float* __restrict__ logits, const unsigned* __restrict__ m_u,
                         const int* __restrict__ dst, float* __restrict__ s, size_t totalEH)
{
    size_t i = (size_t)blockIdx.x * 256 + threadIdx.x;
    if (i >= totalEH) return;
    size_t e = i >> 3; int h = (int)(i & 7);
    int di = dst[e];
    float m = key2f(m_u[(size_t)di * 8 + h]);
    float p = __expf(logits[i] - m);
    logits[i] = p;
    atomicAdd(&s[(size_t)di * 8 + h], p);
}

// 32 lanes/edge, 4 channels each: out[dst] += (p/s[dst]) * v[src]
__global__ __launch_bounds__(256)
void attn_agg(const float* __restrict__ p, const float* __restrict__ s,
              const float* __restrict__ v, const int* __restrict__ src,
              const int* __restrict__ dst, float* __restrict__ out, int E)
{
    int tid = threadIdx.x;
    int lane = tid & 31;
    int e = blockIdx.x * 8 + (tid >> 5);
    if (e >= E) return;
    int sj = src[e], di = dst[e];
    int ch = lane * 4, head = lane >> 2;
    float attn = p[(size_t)e * 8 + head] / s[(size_t)di * 8 + head];
    float4 vv = *(const float4*)(v + (size_t)sj * 128 + ch);
    float* d = out + (size_t)di * 128 + ch;
    atomicAdd(d + 0, attn * vv.x); atomicAdd(d + 1, attn * vv.y);
    atomicAdd(d + 2, attn * vv.z); atomicAdd(d + 3, attn * vv.w);
}

// ---------------------------------------------------------------------------
extern "C" void kernel_launch(void* const* d_in, const int* in_sizes, int n_in,
                              void* d_out, int out_size, void* d_ws, size_t ws_size,
                              hipStream_t stream)
{
    const float* x          = (const float*)d_in[0];
    const int*   edge_index = (const int*)d_in[1];
    const float* edge_attr  = (const float*)d_in[2];
    const int*   dag        = (const int*)d_in[3];
    const float* W_v   = (const float*)d_in[4];
    const float* W_qk  = (const float*)d_in[5];
    const float* gcn_W = (const float*)d_in[6];
    const float* gcn_b = (const float*)d_in[7];
    const float* gcn_root = (const float*)d_in[8];
    const float* gcn_We   = (const float*)d_in[9];
    const float* gcn_be   = (const float*)d_in[10];
    const float* bn_gamma = (const float*)d_in[11];
    const float* bn_beta  = (const float*)d_in[12];
    const float* W_se  = (const float*)d_in[13];
    const float* b_se  = (const float*)d_in[14];
    const float* W_out = (const float*)d_in[15];
    const float* b_out = (const float*)d_in[16];

    const int N = in_sizes[0] / 128;      // 50000 (divisible by 16)
    const int E = in_sizes[2] / 16;       // 800000 (divisible by 32)
    const int* row = edge_index;
    const int* col = edge_index + E;
    const int* src = dag;
    const int* dst = dag + E;

    const size_t n = (size_t)N;
    float* ws      = (float*)d_ws;
    float* xcat    = ws;                      // N x 512  [x | h1 | h2 | h3]
    float* hw      = xcat    + 512 * n;       // N x 128
    float* agg     = hw      + 128 * n;       // N x 128 (also attn output)
    float* xstruct = agg     + 128 * n;       // N x 128
    float* vbuf    = xstruct + 128 * n;       // N x 128
    float* qk      = vbuf    + 128 * n;       // N x 256
    float* dis     = qk      + 256 * n;       // N
    float* invdeg  = dis     + n;             // N
    float* deg     = invdeg  + n;             // N
    unsigned* m_u  = (unsigned*)(deg + n);    // N x 8
    float* sden    = (float*)m_u + 8 * n;     // N x 8
    float* bnsum   = sden    + 8 * n;         // 512
    float* bnsumsq = bnsum   + 512;           // 512
    float* bnscale = bnsumsq + 512;           // 512
    float* bnshift = bnscale + 512;           // 512
    float* plog    = bnshift + 512;           // E x 8

    auto cdiv = [](size_t a, size_t b) { return (unsigned)((a + b - 1) / b); };
    const dim3 gblk(32, 8);

    // degrees (from undirected-GCN edge_index row)
    fill_f32<<<cdiv(n, 256), 256, 0, stream>>>(deg, 0.f, n);
    deg_count<<<cdiv(E, 256), 256, 0, stream>>>(row, deg, E);
    deg_final<<<cdiv(n, 256), 256, 0, stream>>>(deg, dis, invdeg, N);

    // xcat block 0 = x
    copy_x_to_xcat<<<cdiv(128 * n, 256), 256, 0, stream>>>(x, xcat, 128 * n);

    // 3 GCN layers
    for (int l = 0; l < 3; ++l) {
        gemm_wmma_f32<<<N / 16, gblk, 0, stream>>>(
            xcat + l * 128, 512, gcn_W + (size_t)l * 128 * 128, 128,
            gcn_b + l * 128, hw, 128, 128, 128);
        fill_f32<<<cdiv(128 * n, 256), 256, 0, stream>>>(agg, 0.f, 128 * n);
        gcn_scatter<<<E / 8, 256, 0, stream>>>(
            hw, edge_attr, gcn_We + (size_t)l * 16 * 128, gcn_be + l * 128,
            row, col, dis, agg, E);
        gcn_combine<<<cdiv(128 * n, 256), 256, 0, stream>>>(
            hw, agg, gcn_root + l * 128, invdeg, xcat + (l + 1) * 128, 512, 128 * n);
    }

    // batch norm over xcat (N x 512)
    fill_f32<<<cdiv(1024, 256), 256, 0, stream>>>(bnsum, 0.f, 1024); // sum+sumsq
    bn_stats<<<2048, 256, 0, stream>>>(xcat, bnsum, bnsumsq, 512 * n);
    bn_final<<<2, 256, 0, stream>>>(bnsum, bnsumsq, bn_gamma, bn_beta,
                                    bnscale, bnshift, (float)N);
    bn_norm<<<cdiv(512 * n, 256), 256, 0, stream>>>(xcat, bnscale, bnshift, 512 * n);

    // projections
    gemm_wmma_f32<<<N / 16, gblk, 0, stream>>>(xcat, 512, W_se, 128, b_se,
                                               xstruct, 128, 512, 128);
    gemm_wmma_f32<<<N / 16, gblk, 0, stream>>>(x, 128, W_v, 128, nullptr,
                                               vbuf, 128, 128, 128);
    gemm_wmma_f32<<<N / 16, gblk, 0, stream>>>(xstruct, 128, W_qk, 256, nullptr,
                                               qk, 256, 128, 256);

    // edge softmax attention over dag edges
    fill_u32<<<cdiv(8 * n, 256), 256, 0, stream>>>(m_u, 0x007FFFFFu, 8 * n); // key(-inf)
    fill_f32<<<cdiv(8 * n, 256), 256, 0, stream>>>(sden, 0.f, 8 * n);
    attn_logits<<<E / 32, 256, 0, stream>>>(qk, src, dst, plog, m_u, E);
    attn_exp<<<cdiv((size_t)E * 8, 256), 256, 0, stream>>>(plog, m_u, dst, sden, (size_t)E * 8);
    fill_f32<<<cdiv(128 * n, 256), 256, 0, stream>>>(agg, 0.f, 128 * n);
    attn_agg<<<E / 8, 256, 0, stream>>>(plog, sden, vbuf, src, dst, agg, E);

    // out = attn_out @ W_out + b_out
    gemm_wmma_f32<<<N / 16, gblk, 0, stream>>>(agg, 128, W_out, 128, b_out,
                                               (float*)d_out, 128, 128, 128);
}